// FullyConnectedNetworkClassifier_77558519431587
// MI455X (gfx1250) — compile-verified
//
#include <hip/hip_runtime.h>

typedef __attribute__((ext_vector_type(2))) float v2f;
typedef __attribute__((ext_vector_type(8))) float v8f;

constexpr int kD      = 128;   // embedding dim
constexpr int kB      = 1024;  // batch
constexpr int kV      = 20;    // visits
constexpr int kS      = 30;    // set size
constexpr int kVS     = kV * kS;
constexpr int kIn0    = 2 * kD;  // 256
constexpr int kH0     = 256;
constexpr int kH1     = 128;
constexpr int kNClass = 20;

// ---------------------------------------------------------------------------
// Kernel 1: embedding gather + mean(S) + sum(V) pooling -> x[B, 2D]
// One block per batch element, 256 threads: t<128 -> diag dim t, t>=128 ->
// proc dim t-128. Indices are block-uniform -> staged in LDS once. Each
// gather row read is 128 consecutive lanes * 4B = one full coalesced line.
// ---------------------------------------------------------------------------
__global__ __launch_bounds__(256) void pool_kernel(
    const int* __restrict__ diag_idx, const int* __restrict__ proc_idx,
    const float* __restrict__ emb, float* __restrict__ x) {
  __shared__ int sidx[2 * kVS];
  const int b = blockIdx.x;
  const int t = threadIdx.x;

  for (int i = t; i < kVS; i += 256) {
    sidx[i]       = diag_idx[(size_t)b * kVS + i];
    sidx[kVS + i] = proc_idx[(size_t)b * kVS + i];
  }
  __syncthreads();

  const int table = t >> 7;   // 0 = diag, 1 = proc
  const int dim   = t & 127;
  const int* idx  = sidx + table * kVS;

  float acc = 0.0f;
  for (int i = 0; i < kVS; ++i) {
    // hide random-access latency: prefetch a few rows ahead (L2-resident table)
    if (i + 6 < kVS)
      __builtin_prefetch(emb + (size_t)idx[i + 6] * kD + dim, 0, 1);
    acc += emb[(size_t)idx[i] * kD + dim];
  }
  // mean over S then sum over V == (1/S) * sum over all V*S entries
  x[(size_t)b * kIn0 + table * kD + dim] = acc * (1.0f / kS);
}

// ---------------------------------------------------------------------------
// Kernel 2: fused 3-layer MLP on 16-row tiles using V_WMMA_F32_16X16X4_F32.
//   layer0: [16,256] x W0[256,256]^T + b0 -> relu -> LDS
//   layer1: [16,256] x W1[128,256]^T + b1 -> relu -> LDS
//   layer2: [16,128] x W2[20,128]^T  + b2 -> out
// 128 threads = 4 waves; n-tiles distributed across waves; WMMA executes
// with full EXEC in every wave that runs it.
//
// Fragment layouts per CDNA5 ISA (wave32):
//   A (16x4 f32):  lanes 0-15: v0=A[l][k], v1=A[l][k+1]
//                  lanes 16-31: v0=A[l][k+2], v1=A[l][k+3]
//   B (4x16 f32):  lanes 0-15: v0=B[k][l]=W[n0+l][k],   v1=W[n0+l][k+1]
//                  lanes 16-31: v0=W[n0+l][k+2],        v1=W[n0+l][k+3]
//   C/D (16x16):   VGPR r: lanes 0-15 -> D[r][n0+l]; lanes 16-31 -> D[r+8][n0+l]
// ---------------------------------------------------------------------------
__device__ __forceinline__ v8f wmma_f32(v2f a, v2f b, v8f c) {
  return __builtin_amdgcn_wmma_f32_16x16x4_f32(
      /*neg_a=*/false, a, /*neg_b=*/false, b,
      /*c_mod=*/(short)0, c, /*reuse_a=*/false, /*reuse_b=*/false);
}

__global__ __launch_bounds__(128) void mlp_kernel(
    const float* __restrict__ x,
    const float* __restrict__ W0, const float* __restrict__ b0,
    const float* __restrict__ W1, const float* __restrict__ b1,
    const float* __restrict__ W2, const float* __restrict__ b2,
    float* __restrict__ out) {
  __shared__ float bufA[16 * 256];  // layer input; later reused for h1
  __shared__ float bufB[16 * 256];  // h0 (relu'd layer-0 output)

  const int row0 = blockIdx.x * 16;
  const int tid  = threadIdx.x;
  const int wave = tid >> 5;
  const int lane = tid & 31;
  const int half = lane >> 4;   // 0: K0/K1 + D rows 0-7, 1: K2/K3 + D rows 8-15
  const int l    = lane & 15;

  // stage X tile [16, 256] into LDS
  for (int i = tid; i < 16 * kIn0; i += 128)
    bufA[i] = x[(size_t)row0 * kIn0 + i];
  __syncthreads();

  // ---- layer 0: 16 n-tiles over 4 waves, K = 256 ----
  for (int nt = wave; nt < kH0 / 16; nt += 4) {
    const int n0 = nt * 16;
    v8f c = {};
    const float* arow = bufA + l * kIn0 + 2 * half;
    const float* wrow = W0 + (size_t)(n0 + l) * kIn0 + 2 * half;
#pragma unroll 4
    for (int k = 0; k < kIn0; k += 4) {
      v2f a, bm;
      a.x  = arow[k]; a.y  = arow[k + 1];
      bm.x = wrow[k]; bm.y = wrow[k + 1];
      c = wmma_f32(a, bm, c);
    }
    const float bias = b0[n0 + l];
#pragma unroll
    for (int r = 0; r < 8; ++r) {
      float v = c[r] + bias;
      bufB[(r + 8 * half) * kH0 + n0 + l] = v > 0.0f ? v : 0.0f;  // relu
    }
  }
  __syncthreads();

  // ---- layer 1: 8 n-tiles over 4 waves, K = 256; output into bufA ----
  for (int nt = wave; nt < kH1 / 16; nt += 4) {
    const int n0 = nt * 16;
    v8f c = {};
    const float* arow = bufB + l * kH0 + 2 * half;
    const float* wrow = W1 + (size_t)(n0 + l) * kH0 + 2 * half;
#pragma unroll 4
    for (int k = 0; k < kH0; k += 4) {
      v2f a, bm;
      a.x  = arow[k]; a.y  = arow[k + 1];
      bm.x = wrow[k]; bm.y = wrow[k + 1];
      c = wmma_f32(a, bm, c);
    }
    const float bias = b1[n0 + l];
#pragma unroll
    for (int r = 0; r < 8; ++r) {
      float v = c[r] + bias;
      bufA[(r + 8 * half) * kH1 + n0 + l] = v > 0.0f ? v : 0.0f;  // relu
    }
  }
  __syncthreads();

  // ---- layer 2: N = 20 -> 2 n-tiles on waves 0/1, K = 128 ----
  if (wave < 2) {
    const int n0 = wave * 16;
    const int n  = n0 + l;
    const int nc = n < kNClass ? n : kNClass - 1;  // clamp weight row for pad lanes
    v8f c = {};
    const float* arow = bufA + l * kH1 + 2 * half;
    const float* wrow = W2 + (size_t)nc * kH1 + 2 * half;
#pragma unroll 4
    for (int k = 0; k < kH1; k += 4) {
      v2f a, bm;
      a.x  = arow[k]; a.y  = arow[k + 1];
      bm.x = wrow[k]; bm.y = wrow[k + 1];
      c = wmma_f32(a, bm, c);
    }
    if (n < kNClass) {  // divergence only after the WMMA loop
      const float bias = b2[n];
#pragma unroll
      for (int r = 0; r < 8; ++r)
        out[(size_t)(row0 + r + 8 * half) * kNClass + n] = c[r] + bias;
    }
  }
}

// ---------------------------------------------------------------------------
extern "C" void kernel_launch(void* const* d_in, const int* in_sizes, int n_in,
                              void* d_out, int out_size, void* d_ws, size_t ws_size,
                              hipStream_t stream) {
  const int*   diag = (const int*)d_in[0];
  const int*   proc = (const int*)d_in[1];
  const float* emb  = (const float*)d_in[2];
  const float* W0   = (const float*)d_in[3];
  const float* b0   = (const float*)d_in[4];
  const float* W1   = (const float*)d_in[5];
  const float* b1   = (const float*)d_in[6];
  const float* W2   = (const float*)d_in[7];
  const float* b2   = (const float*)d_in[8];

  float* x   = (float*)d_ws;   // [1024, 256] fp32 = 1 MB scratch
  float* out = (float*)d_out;  // [1024, 20] fp32

  pool_kernel<<<kB, 256, 0, stream>>>(diag, proc, emb, x);
  mlp_kernel<<<kB / 16, 128, 0, stream>>>(x, W0, b0, W1, b1, W2, b2, out);
}